// Group_21492016350013
// MI455X (gfx1250) — compile-verified
//
#include <hip/hip_runtime.h>

// Problem constants (from reference): xyz (32, 8192, 3) f32
#define B_   32
#define N_   8192
#define G_   512      // NUM_GROUP
#define M_   32       // GROUP_SIZE

typedef __attribute__((ext_vector_type(2))) float v2f;
typedef __attribute__((ext_vector_type(8))) float v8f;
typedef unsigned int u32;
typedef __attribute__((ext_vector_type(4))) u32 v4u;
typedef __attribute__((ext_vector_type(8))) int v8i;

// ---------------- wave32 (d, idx) reductions, lexicographic tie-break ----------------
__device__ __forceinline__ void wave_argmax(float& v, int& i) {
#pragma unroll
  for (int off = 16; off >= 1; off >>= 1) {
    float ov = __shfl_xor(v, off, 32);
    int   oi = __shfl_xor(i, off, 32);
    if (ov > v || (ov == v && oi < i)) { v = ov; i = oi; }
  }
}
__device__ __forceinline__ void wave_argmin(float& v, int& i) {
#pragma unroll
  for (int off = 16; off >= 1; off >>= 1) {
    float ov = __shfl_xor(v, off, 32);
    int   oi = __shfl_xor(i, off, 32);
    if (ov < v || (ov == v && oi < i)) { v = ov; i = oi; }
  }
}

// =====================================================================================
// Kernel 1: farthest point sampling. One block per batch. The whole 96 KB xyz batch is
// staged into LDS by ONE Tensor Data Mover descriptor (1D tile of 24576 dwords,
// CDNA5 ISA), waited with s_wait_tensorcnt. min_d (32 KB) also
// LDS-resident. 512 serial steps, wave32 shuffle reductions. Writes the 512 centers
// directly to the center output region.
// LDS reads stride 3 dwords per lane -> gcd(3,64)=1 -> bank-conflict-free.
// =====================================================================================
#define FPS_THREADS 1024
#define PTS_PER_THREAD (N_ / FPS_THREADS)  // 8

__global__ __launch_bounds__(FPS_THREADS)
void fps_kernel(const float* __restrict__ xyz, float* __restrict__ out_center) {
  __shared__ float sxyz[N_ * 3];     // interleaved (N,3), filled by TDM
  __shared__ float smind[N_];
  __shared__ float red_v[32];
  __shared__ int   red_i[32];
  __shared__ int   s_far;

  const int b    = blockIdx.x;
  const int tid  = threadIdx.x;
  const int lane = tid & 31;
  const int wave = tid >> 5;
  const float* base = xyz + (size_t)b * N_ * 3;

  for (int p = tid; p < N_; p += FPS_THREADS) smind[p] = 1e10f;
  if (tid == 0) s_far = 0;

  if (wave == 0) {
    // ---- Tensor DMA: contiguous 1D copy, 24576 dwords global -> LDS ----
    const u32 NW = (u32)(N_ * 3);                       // 24576 elements (4B each)
    unsigned long long ga = (unsigned long long)(const void*)base;
    u32 lds = (u32)(unsigned long long)(void*)&sxyz[0]; // low 32 bits = LDS offset
    v4u g0;
    g0.x = 1u;                                          // count=1 (valid), user mode
    g0.y = lds;                                         // lds_addr
    g0.z = (u32)(ga & 0xffffffffu);                     // global_addr[31:0]
    g0.w = (u32)((ga >> 32) & 0x1ffffffu) | (2u << 30); // global_addr[56:32] | type=2
    v8i g1;
    g1[0] = (int)(2u << 16);                 // data_size=2 (4 bytes)
    g1[1] = (int)((NW & 0xffffu) << 16);     // tensor_dim0[15:0]
    g1[2] = (int)(((NW >> 16) & 0xffffu) | (1u << 16)); // tensor_dim0[31:16], tensor_dim1=1
    g1[3] = (int)((NW & 0xffffu) << 16);     // tile_dim0 = NW (16-bit ok: 24576)
    g1[4] = 0;                               // tile_dim1=0 (unused), tile_dim2=0
    g1[5] = (int)NW;                         // tensor_dim0_stride[31:0]
    g1[6] = 0;                               // stride hi, tensor_dim1_stride lo
    g1[7] = 0;
    asm volatile("tensor_load_to_lds %0, %1" :: "s"(g0), "s"(g1) : "memory");
    __builtin_amdgcn_s_wait_tensorcnt(0);
  }
  __syncthreads();

  int farthest = 0;
  float* outc = out_center + (size_t)b * G_ * 3;

  for (int it = 0; it < G_; ++it) {
    // reference emits the *current* farthest as sample `it`, then advances
    float cx = sxyz[3 * farthest + 0];
    float cy = sxyz[3 * farthest + 1];
    float cz = sxyz[3 * farthest + 2];
    if (tid == 0) { outc[it * 3 + 0] = cx; outc[it * 3 + 1] = cy; outc[it * 3 + 2] = cz; }

    float bv = -1.0f; int bi = 0x7fffffff;
#pragma unroll
    for (int k = 0; k < PTS_PER_THREAD; ++k) {
      int p = tid + k * FPS_THREADS;
      float dx = sxyz[3 * p + 0] - cx;
      float dy = sxyz[3 * p + 1] - cy;
      float dz = sxyz[3 * p + 2] - cz;
      float d = dx * dx + dy * dy + dz * dz;
      float m = fminf(smind[p], d);
      smind[p] = m;
      if (m > bv) { bv = m; bi = p; }  // strided p increasing -> first-max kept
    }
    wave_argmax(bv, bi);
    if (lane == 0) { red_v[wave] = bv; red_i[wave] = bi; }
    __syncthreads();
    if (wave == 0) {
      float v = red_v[lane]; int i = red_i[lane];
      wave_argmax(v, i);
      if (lane == 0) s_far = i;
    }
    __syncthreads();
    farthest = s_far;
  }
}

// =====================================================================================
// Kernel 2: KNN + gather + recenter.
// One block = 16 centers of one batch (WMMA M dimension). d2 computed via
// V_WMMA_F32_16X16X4_F32:  A row = (cx, cy, cz, |c|^2), B col = (-2x, -2y, -2z, 1)
//   => A*B = |c|^2 - 2 c.x ; |x|^2 added when the tile is spilled to LDS.
// d2 processed in 16 x 2048 LDS chunks (128 KB). Per-center top-32 lives in wave
// registers (lane j = j-th nearest), sorted-insert via ballot + shfl_up.
// Next chunk's xyz prefetched (global_prefetch_b8) while this chunk is consumed.
// =====================================================================================
#define CHUNK 2048
#define NCHUNK (N_ / CHUNK)                 // 4
#define K2_THREADS 256                      // 8 waves
#define K2_WAVES 8
#define TILES_PER_CHUNK (CHUNK / 16)        // 128

__global__ __launch_bounds__(K2_THREADS)
void knn_kernel(const float* __restrict__ xyz, const float* __restrict__ centers,
                float* __restrict__ out_nb) {
  __shared__ float d2[16 * CHUNK];

  const int tid  = threadIdx.x;
  const int lane = tid & 31;
  const int wave = tid >> 5;
  const int b    = blockIdx.x >> 5;   // G_/16 == 32 center-tiles per batch
  const int gt   = blockIdx.x & 31;
  const int g0   = gt * 16;

  const float* xb = xyz + (size_t)b * N_ * 3;
  const float* cb = centers + ((size_t)b * G_ + g0) * 3;

  // A operand (16x4 f32): lanes 0-15 hold (K0,K1) of row M=lane; lanes 16-31 (K2,K3).
  const int half = lane >> 4;
  const int l15  = lane & 15;
  float c0 = cb[l15 * 3 + 0], c1 = cb[l15 * 3 + 1], c2v = cb[l15 * 3 + 2];
  float cn = c0 * c0 + c1 * c1 + c2v * c2v;
  v2f amat;
  amat.x = half ? c2v : c0;
  amat.y = half ? cn  : c1;

  // running top-32 for the wave's two centers (m = wave*2 + s); lane j = j-th smallest
  float rd0 = 3e38f, rd1 = 3e38f;
  int   ri0 = 0x7fffffff, ri1 = 0x7fffffff;

  for (int ch = 0; ch < NCHUNK; ++ch) {
    const int nbase = ch * CHUNK;
    __syncthreads();  // previous chunk's selection done before overwrite

    // prefetch next chunk's xyz (24 KB) into cache while we crunch this one
    if (ch + 1 < NCHUNK) {
      const char* nf = (const char*)(xb + (size_t)(nbase + CHUNK) * 3);
      __builtin_prefetch(nf + tid * 96, 0, 1);
    }

    // ---- WMMA phase (no divergence: EXEC all ones) ----
    for (int t = wave; t < TILES_PER_CHUNK; t += K2_WAVES) {
      int ncol = nbase + t * 16 + l15;
      const float* p = xb + (size_t)ncol * 3;
      float px = p[0], py = p[1], pz = p[2];
      float xn = px * px + py * py + pz * pz;
      // B operand (4x16 f32): lanes 0-15 hold (K0,K1) of col N=lane; lanes 16-31 (K2,K3)
      v2f bmat;
      bmat.x = half ? (-2.0f * pz) : (-2.0f * px);
      bmat.y = half ? 1.0f         : (-2.0f * py);
      v8f acc = {};
      acc = __builtin_amdgcn_wmma_f32_16x16x4_f32(
          /*neg_a=*/false, amat, /*neg_b=*/false, bmat,
          /*c_mod=*/(short)0, acc, /*reuse_a=*/false, /*reuse_b=*/false);
      // C/D layout: VGPR v -> M=v (lanes 0-15) or M=v+8 (lanes 16-31), N = lane&15
#pragma unroll
      for (int v = 0; v < 8; ++v) {
        int row = v + half * 8;
        d2[row * CHUNK + t * 16 + l15] = acc[v] + xn;
      }
    }
    __syncthreads();

    // ---- selection phase: wave handles centers m = wave*2 .. wave*2+1 ----
    for (int s = 0; s < 2; ++s) {
      int m = wave * 2 + s;
      const float* row = &d2[m * CHUNK];
      float rd = s ? rd1 : rd0;
      int   ri = s ? ri1 : ri0;
      float last_d = -3e38f; int last_i = -1;  // extracted-so-far fence (lexicographic)
      for (int t2 = 0; t2 < M_; ++t2) {
        float bd = 3e38f; int bi = 0x7fffffff;
        for (int k = lane; k < CHUNK; k += 32) {
          float v  = row[k];
          int   gi = nbase + k;
          bool newer  = (v > last_d) || (v == last_d && gi > last_i);
          bool better = (v < bd)     || (v == bd     && gi < bi);
          if (newer && better) { bd = v; bi = gi; }
        }
        wave_argmin(bd, bi);
        float wd = __shfl(rd, 31, 32); int wi = __shfl(ri, 31, 32);
        if (!((bd < wd) || (bd == wd && bi < wi))) break;  // can't enter top-32
        // parallel sorted insert
        bool before = (bd < rd) || (bd == rd && bi < ri);
        unsigned long long msk = __ballot(before);
        int pos = (int)__builtin_ctzll(msk);  // msk != 0: lane 31 flagged
        float pd = __shfl_up(rd, 1, 32);
        int   pi = __shfl_up(ri, 1, 32);
        if (lane > pos)       { rd = pd; ri = pi; }
        else if (lane == pos) { rd = bd; ri = bi; }
        last_d = bd; last_i = bi;
      }
      if (s) { rd1 = rd; ri1 = ri; } else { rd0 = rd; ri0 = ri; }
    }
  }

  // ---- gather + recenter: lane j writes neighbor j of each of the wave's 2 centers ----
#pragma unroll
  for (int s = 0; s < 2; ++s) {
    int m = wave * 2 + s;
    int g = g0 + m;
    int ri = s ? ri1 : ri0;
    float cx = cb[m * 3 + 0], cy = cb[m * 3 + 1], cz = cb[m * 3 + 2];
    const float* p = xb + (size_t)ri * 3;
    float* o = out_nb + (((size_t)b * G_ + g) * M_ + lane) * 3;
    o[0] = p[0] - cx; o[1] = p[1] - cy; o[2] = p[2] - cz;
  }
}

// =====================================================================================
extern "C" void kernel_launch(void* const* d_in, const int* in_sizes, int n_in,
                              void* d_out, int out_size, void* d_ws, size_t ws_size,
                              hipStream_t stream) {
  (void)in_sizes; (void)n_in; (void)out_size; (void)d_ws; (void)ws_size;
  const float* xyz = (const float*)d_in[0];
  float* out        = (float*)d_out;
  float* out_nb     = out;                                   // (B, G, 32, 3)
  float* out_center = out + (size_t)B_ * G_ * M_ * 3;        // (B, G, 3)

  fps_kernel<<<B_, FPS_THREADS, 0, stream>>>(xyz, out_center);
  knn_kernel<<<B_ * (G_ / 16), K2_THREADS, 0, stream>>>(xyz, out_center, out_nb);
}